// GCN_x_86732569575635
// MI455X (gfx1250) — compile-verified
//
#include <hip/hip_runtime.h>
#include <hip/hip_bf16.h>

typedef __attribute__((ext_vector_type(2))) float v2f;
typedef __attribute__((ext_vector_type(4))) float v4f;
typedef __attribute__((ext_vector_type(8))) float v8f;

#define NNODES  100000
#define MPAD    100096          // multiple of 128
#define NEDGES  1600000
#define NGRAPHS 512

// -------------------------------------------------------------------------
// utility kernels
// -------------------------------------------------------------------------
__global__ void zero_kernel(float* __restrict__ p, long long n) {
  long long i = (long long)blockIdx.x * blockDim.x + threadIdx.x;
  if (i < n) p[i] = 0.0f;
}

__global__ void deg_count_kernel(const int* __restrict__ dst, float* __restrict__ deg, int E) {
  int e = blockIdx.x * blockDim.x + threadIdx.x;
  if (e < E) atomicAdd(&deg[dst[e]], 1.0f);
}

__global__ void dinv_kernel(float* __restrict__ deg, int n) {
  int i = blockIdx.x * blockDim.x + threadIdx.x;
  if (i < n) deg[i] = rsqrtf(deg[i] + 1.0f);
}

// pad x[NNODES,3] -> xp[MPAD,4]
__global__ void pad_x_kernel(const float* __restrict__ x, float* __restrict__ xp) {
  long long i = (long long)blockIdx.x * blockDim.x + threadIdx.x;
  if (i >= (long long)MPAD * 4) return;
  int node = (int)(i >> 2), j = (int)(i & 3);
  xp[i] = (node < NNODES && j < 3) ? x[(size_t)node * 3 + j] : 0.0f;
}

// pad W1[3,128] -> W1p[4,128]
__global__ void pad_w1_kernel(const float* __restrict__ w, float* __restrict__ wp) {
  int i = blockIdx.x * blockDim.x + threadIdx.x;
  if (i >= 4 * 128) return;
  int k = i >> 7;
  wp[i] = (k < 3) ? w[i] : 0.0f;
}

// pad Wl2[32,9] -> Wl2p[32,16], bl2[9] -> bl2p[16]
__global__ void pad_wl2_kernel(const float* __restrict__ w, const float* __restrict__ b,
                               float* __restrict__ wp, float* __restrict__ bp) {
  int i = blockIdx.x * blockDim.x + threadIdx.x;
  if (i >= 32 * 16) return;
  int k = i >> 4, n = i & 15;
  wp[i] = (n < 9) ? w[k * 9 + n] : 0.0f;
  if (i < 16) bp[i] = (i < 9) ? b[i] : 0.0f;
}

// o16[512,16] -> out[512,9]
__global__ void copy_out_kernel(const float* __restrict__ o16, float* __restrict__ out) {
  int i = blockIdx.x * blockDim.x + threadIdx.x;
  if (i >= NGRAPHS * 9) return;
  int g = i / 9, j = i - g * 9;
  out[i] = o16[g * 16 + j];
}

// -------------------------------------------------------------------------
// WMMA f32 GEMM:  C[M,N] = act(A[M,K]) @ W[K,N] (+bias)
// K%4==0, N%16==0, M%128==0.  Block = 256 threads = 8 waves = 128 rows.
// W is staged once per block into LDS, pair-interleaved so the per-lane
// B operand (W[kk][n], W[kk+1][n]) is one aligned ds_load_b64.
// -------------------------------------------------------------------------
template <int K, int N, bool RELU, bool BIAS>
__global__ __launch_bounds__(256) void gemm_wmma(
    const float* __restrict__ A, const float* __restrict__ W,
    const float* __restrict__ bias, float* __restrict__ C) {
  __shared__ float lw[K * N];   // layout: ((k>>1)*N + n)*2 + (k&1)

  const int tid = threadIdx.x;
  // ---- stage W into LDS (pair-interleaved), fully vectorized ----
  constexpr int QUADS = (K / 2) * N / 4;     // groups of 4 pairs
  for (int q = tid; q < QUADS; q += 256) {
    int t = (q * 4) / N;                     // pair row (k = 2t)
    int n = (q * 4) % N;
    int k = t * 2;
    v4f w0 = *(const v4f*)(W + (size_t)k * N + n);        // global_load_b128
    v4f w1 = *(const v4f*)(W + (size_t)(k + 1) * N + n);  // global_load_b128
    v4f lo, hi;
    lo.x = w0.x; lo.y = w1.x; lo.z = w0.y; lo.w = w1.y;
    hi.x = w0.z; hi.y = w1.z; hi.z = w0.w; hi.w = w1.w;
    float* d = lw + ((size_t)t * N + n) * 2;
    *(v4f*)(d)     = lo;                                  // ds_store_b128
    *(v4f*)(d + 4) = hi;                                  // ds_store_b128
  }
  __syncthreads();

  const int lane = tid & 31;
  const int wave = tid >> 5;
  const int half = lane >> 4;               // selects K pair (ISA A/B layout)
  const int l16  = lane & 15;
  const int row  = blockIdx.x * 128 + wave * 16 + l16;
  const float* __restrict__ Arow = A + (size_t)row * K + 2 * half;

  for (int n_base = 0; n_base < N; n_base += 16) {
    const int n = n_base + l16;
    float binit = BIAS ? bias[n] : 0.0f;
    v8f acc;
#pragma unroll
    for (int r = 0; r < 8; ++r) acc[r] = binit;

#pragma unroll
    for (int k = 0; k < K; k += 4) {
      v2f av = *(const v2f*)(Arow + k);                   // global_load_b64
      if (RELU) { av.x = fmaxf(av.x, 0.0f); av.y = fmaxf(av.y, 0.0f); }
      const int kk = k + 2 * half;
      v2f bv = *(const v2f*)(lw + ((size_t)(kk >> 1) * N + n) * 2);  // ds_load_b64
      acc = __builtin_amdgcn_wmma_f32_16x16x4_f32(
          false, av, false, bv, (short)0, acc, false, false);
    }

    float* __restrict__ Cp = C + (size_t)(blockIdx.x * 128 + wave * 16 + half * 8) * N + n;
#pragma unroll
    for (int r = 0; r < 8; ++r) Cp[(size_t)r * N] = acc[r];
  }
}

// -------------------------------------------------------------------------
// out[i,f] = h[i,f]*dinv[i]^2 + bias[f]; pad rows -> 0 (vectorized x4)
// -------------------------------------------------------------------------
__global__ void selfloop_bias_kernel(const float* __restrict__ h,
                                     const float* __restrict__ dinv,
                                     const float* __restrict__ bias,
                                     float* __restrict__ out,
                                     int F, int lgF4) {
  long long i = (long long)blockIdx.x * blockDim.x + threadIdx.x;
  if (i >= ((long long)MPAD << lgF4)) return;
  int node = (int)(i >> lgF4);
  int f    = (int)(i & ((1 << lgF4) - 1)) * 4;
  v4f o;
  if (node < NNODES) {
    float di = dinv[node];
    float s  = di * di;
    v4f hv = *(const v4f*)(h + (size_t)node * F + f);
    v4f bv = *(const v4f*)(bias + f);
    o.x = hv.x * s + bv.x; o.y = hv.y * s + bv.y;
    o.z = hv.z * s + bv.z; o.w = hv.w * s + bv.w;
  } else {
    o.x = o.y = o.z = o.w = 0.0f;
  }
  *(v4f*)(out + (size_t)node * F + f) = o;
}

// -------------------------------------------------------------------------
// out[dst,f..f+3] += h[src,f..f+3]*dinv[src]*dinv[dst]  (L2-resident atomics)
// -------------------------------------------------------------------------
__global__ void edge_scatter_kernel(const float* __restrict__ h,
                                    const int* __restrict__ src,
                                    const int* __restrict__ dst,
                                    const float* __restrict__ dinv,
                                    float* __restrict__ out,
                                    int E, int F, int lgF4) {
  long long i = (long long)blockIdx.x * blockDim.x + threadIdx.x;
  if (i >= ((long long)E << lgF4)) return;
  int e = (int)(i >> lgF4);
  int f = (int)(i & ((1 << lgF4) - 1)) * 4;
  int s = src[e], d = dst[e];
  float nrm = dinv[s] * dinv[d];
  v4f hv = *(const v4f*)(h + (size_t)s * F + f);
  float* o = out + (size_t)d * F + f;
  atomicAdd(o + 0, hv.x * nrm);
  atomicAdd(o + 1, hv.y * nrm);
  atomicAdd(o + 2, hv.z * nrm);
  atomicAdd(o + 3, hv.w * nrm);
}

// -------------------------------------------------------------------------
// mean pooling over graphs (F = 64)
// -------------------------------------------------------------------------
__global__ void pool_acc_kernel(const float* __restrict__ h,
                                const int* __restrict__ batch,
                                float* __restrict__ pooled,
                                float* __restrict__ cnt) {
  long long i = (long long)blockIdx.x * blockDim.x + threadIdx.x;
  if (i >= (long long)NNODES * 16) return;
  int node = (int)(i >> 4);
  int f    = (int)(i & 15) * 4;
  int g = batch[node];
  v4f hv = *(const v4f*)(h + (size_t)node * 64 + f);
  float* p = pooled + (size_t)g * 64 + f;
  atomicAdd(p + 0, hv.x);
  atomicAdd(p + 1, hv.y);
  atomicAdd(p + 2, hv.z);
  atomicAdd(p + 3, hv.w);
  if (f == 0) atomicAdd(&cnt[g], 1.0f);
}

__global__ void pool_div_kernel(float* __restrict__ pooled,
                                const float* __restrict__ cnt) {
  int i = blockIdx.x * blockDim.x + threadIdx.x;
  if (i < NGRAPHS * 64) pooled[i] /= fmaxf(cnt[i >> 6], 1.0f);
}

// -------------------------------------------------------------------------
// launch
// -------------------------------------------------------------------------
extern "C" void kernel_launch(void* const* d_in, const int* in_sizes, int n_in,
                              void* d_out, int out_size, void* d_ws, size_t ws_size,
                              hipStream_t stream) {
  const float* x   = (const float*)d_in[0];
  const int*   ei  = (const int*)d_in[1];
  const int*   bat = (const int*)d_in[2];
  const float* W1  = (const float*)d_in[3];  const float* b1  = (const float*)d_in[4];
  const float* W2  = (const float*)d_in[5];  const float* b2  = (const float*)d_in[6];
  const float* W3  = (const float*)d_in[7];  const float* b3  = (const float*)d_in[8];
  const float* W4  = (const float*)d_in[9];  const float* b4  = (const float*)d_in[10];
  const float* Wl1 = (const float*)d_in[11]; const float* bl1 = (const float*)d_in[12];
  const float* Wl2 = (const float*)d_in[13]; const float* bl2 = (const float*)d_in[14];
  float* out = (float*)d_out;

  const int* src = ei;
  const int* dst = ei + NEDGES;

  // workspace carve-up (floats); all chunks multiple of 16 floats
  float* ws     = (float*)d_ws;
  float* dinv   = ws;                              // NNODES
  float* xp     = dinv + NNODES;                   // MPAD*4
  float* hT     = xp + (size_t)MPAD * 4;           // MPAD*128
  float* hA     = hT + (size_t)MPAD * 128;         // MPAD*128
  float* pooled = hA + (size_t)MPAD * 128;         // NGRAPHS*64
  float* cnt    = pooled + NGRAPHS * 64;           // NGRAPHS
  float* t1     = cnt + NGRAPHS;                   // NGRAPHS*32
  float* W1p    = t1 + NGRAPHS * 32;               // 4*128
  float* Wl2p   = W1p + 4 * 128;                   // 32*16
  float* bl2p   = Wl2p + 32 * 16;                  // 16
  float* o16    = bl2p + 16;                       // NGRAPHS*16

  const int B = 256;
  auto blks = [](long long n, int b) { return (unsigned)((n + b - 1) / b); };

  // degrees -> dinv
  zero_kernel<<<blks(NNODES, B), B, 0, stream>>>(dinv, NNODES);
  deg_count_kernel<<<blks(NEDGES, B), B, 0, stream>>>(dst, dinv, NEDGES);
  dinv_kernel<<<blks(NNODES, B), B, 0, stream>>>(dinv, NNODES);

  // padding
  pad_x_kernel<<<blks((long long)MPAD * 4, B), B, 0, stream>>>(x, xp);
  pad_w1_kernel<<<blks(4 * 128, B), B, 0, stream>>>(W1, W1p);
  pad_wl2_kernel<<<blks(32 * 16, B), B, 0, stream>>>(Wl2, bl2, Wl2p, bl2p);

  const unsigned GB = MPAD / 128;
  const long long n128 = (long long)MPAD * 32;
  const long long n64  = (long long)MPAD * 16;
  const long long e128 = (long long)NEDGES * 32;
  const long long e64  = (long long)NEDGES * 16;

  // ---- layer 1 ----
  gemm_wmma<4, 128, false, false><<<GB, 256, 0, stream>>>(xp, W1p, nullptr, hT);
  selfloop_bias_kernel<<<blks(n128, B), B, 0, stream>>>(hT, dinv, b1, hA, 128, 5);
  edge_scatter_kernel<<<blks(e128, B), B, 0, stream>>>(hT, src, dst, dinv, hA, NEDGES, 128, 5);

  // ---- layer 2 ----
  gemm_wmma<128, 128, true, false><<<GB, 256, 0, stream>>>(hA, W2, nullptr, hT);
  selfloop_bias_kernel<<<blks(n128, B), B, 0, stream>>>(hT, dinv, b2, hA, 128, 5);
  edge_scatter_kernel<<<blks(e128, B), B, 0, stream>>>(hT, src, dst, dinv, hA, NEDGES, 128, 5);

  // ---- layer 3 ----
  gemm_wmma<128, 128, true, false><<<GB, 256, 0, stream>>>(hA, W3, nullptr, hT);
  selfloop_bias_kernel<<<blks(n128, B), B, 0, stream>>>(hT, dinv, b3, hA, 128, 5);
  edge_scatter_kernel<<<blks(e128, B), B, 0, stream>>>(hT, src, dst, dinv, hA, NEDGES, 128, 5);

  // ---- layer 4 (F = 64, no output relu) ----
  gemm_wmma<128, 64, true, false><<<GB, 256, 0, stream>>>(hA, W4, nullptr, hT);
  selfloop_bias_kernel<<<blks(n64, B), B, 0, stream>>>(hT, dinv, b4, hA, 64, 4);
  edge_scatter_kernel<<<blks(e64, B), B, 0, stream>>>(hT, src, dst, dinv, hA, NEDGES, 64, 4);

  // ---- global mean pool ----
  zero_kernel<<<blks(NGRAPHS * 64, B), B, 0, stream>>>(pooled, NGRAPHS * 64);
  zero_kernel<<<blks(NGRAPHS, B), B, 0, stream>>>(cnt, NGRAPHS);
  pool_acc_kernel<<<blks((long long)NNODES * 16, B), B, 0, stream>>>(hA, bat, pooled, cnt);
  pool_div_kernel<<<blks(NGRAPHS * 64, B), B, 0, stream>>>(pooled, cnt);

  // ---- MLP head ----
  gemm_wmma<64, 32, false, true><<<NGRAPHS / 128, 256, 0, stream>>>(pooled, Wl1, bl1, t1);
  gemm_wmma<32, 16, true, true><<<NGRAPHS / 128, 256, 0, stream>>>(t1, Wl2p, bl2p, o16);
  copy_out_kernel<<<blks(NGRAPHS * 9, B), B, 0, stream>>>(o16, out);
}